// GCNLayer_81466939671129
// MI455X (gfx1250) — compile-verified
//
#include <hip/hip_runtime.h>
#include <hip/hip_bf16.h>

// GCN layer for MI455X (gfx1250, wave32, WMMA).
//   deg -> d_inv -> W transpose -> scatter(msg) -> fused agg+GEMM(relu)
//   GEMM inner loop: v_wmma_f32_16x16x4_f32, B fragments as single b64 loads
//   from transposed W, 2 row-tiles x 4 col-tiles per wave for B reuse.

#define DIN  256
#define DOUT 256

typedef float v2f __attribute__((ext_vector_type(2)));
typedef float v8f __attribute__((ext_vector_type(8)));

// ---------------------------------------------------------------- utilities
__global__ void gcn_zero_kernel(float* __restrict__ p, int n) {
    int i = blockIdx.x * blockDim.x + threadIdx.x;
    if (i < n) p[i] = 0.0f;
}

// deg[src[e]] += 1  (out-degree of src, no self loops — matches reference)
__global__ void gcn_degree_kernel(const int* __restrict__ src,
                                  float* __restrict__ deg, int E) {
    int e = blockIdx.x * blockDim.x + threadIdx.x;
    if (e < E) atomicAdd(&deg[src[e]], 1.0f);
}

__global__ void gcn_dinv_kernel(const float* __restrict__ deg,
                                float* __restrict__ dinv, int n) {
    int i = blockIdx.x * blockDim.x + threadIdx.x;
    if (i < n) dinv[i] = 1.0f / sqrtf(deg[i]);
}

// Wt[n][k] = W[k][n]  (one-time 256KB transpose so B fragments are contiguous)
__global__ void gcn_transpose_kernel(const float* __restrict__ W,
                                     float* __restrict__ Wt) {
    int o = blockIdx.x * blockDim.x + threadIdx.x;   // 65536 elements
    if (o < DIN * DOUT) {
        int n = o >> 8;          // output row  (col of W)
        int k = o & 255;         // output col  (row of W)
        Wt[o] = W[k * DOUT + n];
    }
}

// msg[src] += d_inv[dst] * x[dst]   (64 float4 lanes per edge -> fp32 atomics)
__global__ void gcn_scatter_kernel(const float* __restrict__ x,
                                   const int* __restrict__ src,
                                   const int* __restrict__ dst,
                                   const float* __restrict__ dinv,
                                   float* __restrict__ msg, int E) {
    int tid = blockIdx.x * blockDim.x + threadIdx.x;
    int e = tid >> 6;                 // 64 threads (2 waves) per edge
    if (e >= E) return;
    int q = tid & 63;                 // float4 slot: covers 256 cols
    int s = src[e];
    int d = dst[e];
    float w = dinv[d];
    const float4 xv = ((const float4*)(x + (size_t)d * DIN))[q];
    float* m = msg + (size_t)s * DIN + q * 4;
    atomicAdd(m + 0, w * xv.x);
    atomicAdd(m + 1, w * xv.y);
    atomicAdd(m + 2, w * xv.z);
    atomicAdd(m + 3, w * xv.w);
}

// ---------------------------------------------------------------- fused GEMM
// Block = 128 threads (4 waves). Block computes 32 rows x 256 cols of
// y = relu(agg @ W + b), agg built on the fly into LDS:
//   agg[r][c] = d_inv[r]*msg[r][c] + d_inv[r]^2 * x[r][c]
// Wave w owns col-tiles 4w..4w+3 over BOTH 16-row tiles: 8 accumulators,
// so each B fragment (one ds-free global_load_b64 from Wt) feeds 2 wmmas
// and each A fragment (ds_load_b64) feeds 4 wmmas.
// Fragment layouts per ISA 7.12.2 (f32, 16x16x4):
//   A 16x4:  lanes 0-15 rows 0-15 hold K={0,1}; lanes 16-31 hold K={2,3}
//   B 4x16:  VGPR v = row K=v (+2 upper lane half), cols across lanes
//   C/D:     VGPR v = row v (+8 upper lane half), cols across lanes
__global__ void __launch_bounds__(128)
gcn_gemm_kernel(const float* __restrict__ x,
                const float* __restrict__ msg,
                const float* __restrict__ dinv,
                const float* __restrict__ Wt,    // [DOUT][DIN] (W transposed)
                const float* __restrict__ bias,  // [DOUT]
                float* __restrict__ out,         // [N][DOUT]
                int N) {
    __shared__ float lds[32][DIN + 4];           // +4 floats: conflict-free frags

    const int lane = threadIdx.x & 31;
    const int wave = threadIdx.x >> 5;           // 0..3
    const int rbase = blockIdx.x * 32;

    // ---- stage 32 rows of agg into LDS (each thread: 1 row, 64 cols)
    {
        const int r = threadIdx.x >> 2;          // 0..31
        const int c = (threadIdx.x & 3) * 64;    // 0,64,128,192
        const int g = rbase + r;
        if (g < N) {
            const float di = dinv[g];
            const float di2 = di * di;
            const float4* m4 = (const float4*)(msg + (size_t)g * DIN + c);
            const float4* x4 = (const float4*)(x + (size_t)g * DIN + c);
#pragma unroll
            for (int j = 0; j < 16; ++j) {
                float4 mv = m4[j];
                float4 xv = x4[j];
                float4 av;
                av.x = di * mv.x + di2 * xv.x;
                av.y = di * mv.y + di2 * xv.y;
                av.z = di * mv.z + di2 * xv.z;
                av.w = di * mv.w + di2 * xv.w;
                *(float4*)&lds[r][c + 4 * j] = av;
            }
        } else {
            float4 zv = {0.f, 0.f, 0.f, 0.f};
#pragma unroll
            for (int j = 0; j < 16; ++j)
                *(float4*)&lds[r][c + 4 * j] = zv;
        }
    }
    __syncthreads();

    const int half = lane >> 4;                  // 0: lanes 0-15, 1: lanes 16-31
    const int krow = half << 1;                  // K offset 0 or 2
    const int ln = lane & 15;

    // 4 col-tiles per wave: cols wave*64 + {0,16,32,48} + ln
    const int col0 = wave * 64 + ln;
    const float* wt0 = Wt + (size_t)(col0 +  0) * DIN + krow;
    const float* wt1 = Wt + (size_t)(col0 + 16) * DIN + krow;
    const float* wt2 = Wt + (size_t)(col0 + 32) * DIN + krow;
    const float* wt3 = Wt + (size_t)(col0 + 48) * DIN + krow;

    v8f acc[2][4];
#pragma unroll
    for (int t = 0; t < 2; ++t)
#pragma unroll
        for (int j = 0; j < 4; ++j)
            acc[t][j] = (v8f){0.f, 0.f, 0.f, 0.f, 0.f, 0.f, 0.f, 0.f};

#pragma unroll 2
    for (int k = 0; k < DIN; k += 4) {
        if (k + 16 < DIN)
            __builtin_prefetch(wt0 + k + 16, 0, 3);

        // A fragments for both row tiles (contiguous K pair -> ds_load_b64)
        v2f a0 = *(const v2f*)&lds[ln][k + krow];
        v2f a1 = *(const v2f*)&lds[16 + ln][k + krow];

        // B fragments: contiguous K pair in transposed W -> global_load_b64
        v2f b0 = *(const v2f*)(wt0 + k);
        v2f b1 = *(const v2f*)(wt1 + k);
        v2f b2 = *(const v2f*)(wt2 + k);
        v2f b3 = *(const v2f*)(wt3 + k);

        acc[0][0] = __builtin_amdgcn_wmma_f32_16x16x4_f32(false, a0, false, b0,
                                                (short)0, acc[0][0], false, false);
        acc[1][0] = __builtin_amdgcn_wmma_f32_16x16x4_f32(false, a1, false, b0,
                                                (short)0, acc[1][0], false, false);
        acc[0][1] = __builtin_amdgcn_wmma_f32_16x16x4_f32(false, a0, false, b1,
                                                (short)0, acc[0][1], false, false);
        acc[1][1] = __builtin_amdgcn_wmma_f32_16x16x4_f32(false, a1, false, b1,
                                                (short)0, acc[1][1], false, false);
        acc[0][2] = __builtin_amdgcn_wmma_f32_16x16x4_f32(false, a0, false, b2,
                                                (short)0, acc[0][2], false, false);
        acc[1][2] = __builtin_amdgcn_wmma_f32_16x16x4_f32(false, a1, false, b2,
                                                (short)0, acc[1][2], false, false);
        acc[0][3] = __builtin_amdgcn_wmma_f32_16x16x4_f32(false, a0, false, b3,
                                                (short)0, acc[0][3], false, false);
        acc[1][3] = __builtin_amdgcn_wmma_f32_16x16x4_f32(false, a1, false, b3,
                                                (short)0, acc[1][3], false, false);
    }

    // ---- epilogue: bias + relu, C/D layout per ISA 7.12.2
#pragma unroll
    for (int j = 0; j < 4; ++j) {
        const int col = col0 + j * 16;
        const float bb = bias[col];
#pragma unroll
        for (int t = 0; t < 2; ++t) {
            const int rrow = rbase + t * 16 + (half << 3);
#pragma unroll
            for (int v = 0; v < 8; ++v) {
                const int g = rrow + v;
                if (g < N)
                    out[(size_t)g * DOUT + col] = fmaxf(acc[t][j][v] + bb, 0.0f);
            }
        }
    }
}

// ---------------------------------------------------------------- launcher
extern "C" void kernel_launch(void* const* d_in, const int* in_sizes, int n_in,
                              void* d_out, int out_size, void* d_ws, size_t ws_size,
                              hipStream_t stream) {
    const float* x    = (const float*)d_in[0];
    const int*   eidx = (const int*)d_in[1];     // [2][E]
    const float* W    = (const float*)d_in[2];
    const float* bias = (const float*)d_in[3];
    float* out = (float*)d_out;

    const int N = in_sizes[0] / DIN;
    const int E = in_sizes[1] / 2;
    const int* src = eidx;
    const int* dst = eidx + E;

    float* deg  = (float*)d_ws;                  // [N]
    float* dinv = deg + N;                       // [N]
    float* msg  = dinv + N;                      // [N][DIN]
    float* Wt   = msg + (size_t)N * DIN;         // [DOUT][DIN]

    // 1) zero deg and msg
    gcn_zero_kernel<<<(N + 255) / 256, 256, 0, stream>>>(deg, N);
    gcn_zero_kernel<<<(N * DIN + 255) / 256, 256, 0, stream>>>(msg, N * DIN);

    // 2) out-degree of src
    gcn_degree_kernel<<<(E + 255) / 256, 256, 0, stream>>>(src, deg, E);

    // 3) d_inv = 1/sqrt(deg);  transpose W (independent, same stream)
    gcn_dinv_kernel<<<(N + 255) / 256, 256, 0, stream>>>(deg, dinv, N);
    gcn_transpose_kernel<<<(DIN * DOUT + 255) / 256, 256, 0, stream>>>(W, Wt);

    // 4) scatter messages (2 waves / edge)
    {
        long long threads = (long long)E * 64;
        int blocks = (int)((threads + 255) / 256);
        gcn_scatter_kernel<<<blocks, 256, 0, stream>>>(x, src, dst, dinv, msg, E);
    }

    // 5) fused agg + GEMM + bias + relu  (32 rows per block, row-guarded tail)
    gcn_gemm_kernel<<<(N + 31) / 32, 128, 0, stream>>>(x, msg, dinv, Wt, bias,
                                                       out, N);
}